// MultiScaleVoxelFeatureEncoder_6614249636298
// MI455X (gfx1250) — compile-verified
//
#include <hip/hip_runtime.h>
#include <hip/hip_bf16.h>
#include <math.h>

typedef __attribute__((ext_vector_type(2))) float v2f;
typedef __attribute__((ext_vector_type(8))) float v8f;

// ---------------- utility kernels ----------------

__global__ void zero_f(float* __restrict__ p, size_t n) {
  size_t i = (size_t)blockIdx.x * blockDim.x + threadIdx.x;
  if (i < n) p[i] = 0.0f;
}

__global__ void scatter_add(const float* __restrict__ pts, const int* __restrict__ inv,
                            float* __restrict__ sums, float* __restrict__ cnt, int N) {
  int i = blockIdx.x * blockDim.x + threadIdx.x;
  if (i >= N) return;
  int v = inv[i];
  atomicAdd(&sums[(size_t)v * 4 + 0], pts[(size_t)i * 4 + 0]);
  atomicAdd(&sums[(size_t)v * 4 + 1], pts[(size_t)i * 4 + 1]);
  atomicAdd(&sums[(size_t)v * 4 + 2], pts[(size_t)i * 4 + 2]);
  atomicAdd(&sums[(size_t)v * 4 + 3], pts[(size_t)i * 4 + 3]);
  atomicAdd(&cnt[v], 1.0f);
}

__global__ void mean_div(const float* __restrict__ sums, const float* __restrict__ cnt,
                         float* __restrict__ X, int M) {
  size_t i = (size_t)blockIdx.x * blockDim.x + threadIdx.x;
  if (i >= (size_t)M * 4) return;
  float c = cnt[i >> 2];
  c = c < 1.0f ? 1.0f : c;
  X[i] = sums[i] / c;
}

// ---------------- WMMA f32 GEMM: Y[M,F] = X[M,K] @ W[F,K]^T ----------------
// Each wave: one 16-row tile x 4 column tiles (64 cols). Block = 4 waves =
// 64 rows. A fragment (16x4 f32): lane L holds row L%16, halves K+2*(L/16).
// Rows are CLAMPED (not masked) so the inner loop is branch-free; stores use
// compile-time F so all row/column offsets are immediates, with a branch-free
// fast path for interior tiles.
template <int K, int F>
__global__ void gemm_wmma(const float* __restrict__ X, const float* __restrict__ W,
                          float* __restrict__ Y, int M) {
  const int lane  = threadIdx.x & 31;
  const int wave  = threadIdx.x >> 5;
  const int mtile = blockIdx.x * 4 + wave;
  const int n0    = blockIdx.y * 64 + (lane & 15);  // this lane's column in tile 0
  const int m     = mtile * 16 + (lane & 15);
  const int mc    = m < M ? m : (M - 1);
  const int kh    = (lane >> 4) << 1;  // 0 or 2
  const float* __restrict__ xr = X + (size_t)mc * K + kh;
  const float* __restrict__ w0 = W + (size_t)(n0 +  0) * K + kh;
  const float* __restrict__ w1 = W + (size_t)(n0 + 16) * K + kh;
  const float* __restrict__ w2 = W + (size_t)(n0 + 32) * K + kh;
  const float* __restrict__ w3 = W + (size_t)(n0 + 48) * K + kh;
  v8f c0 = {0.f, 0.f, 0.f, 0.f, 0.f, 0.f, 0.f, 0.f};
  v8f c1 = c0, c2 = c0, c3 = c0;
#pragma unroll 8
  for (int k0 = 0; k0 < K; k0 += 4) {
    v2f a  = *(const v2f*)(xr + k0);
    v2f b0 = *(const v2f*)(w0 + k0);
    v2f b1 = *(const v2f*)(w1 + k0);
    v2f b2 = *(const v2f*)(w2 + k0);
    v2f b3 = *(const v2f*)(w3 + k0);
    c0 = __builtin_amdgcn_wmma_f32_16x16x4_f32(false, a, false, b0, (short)0, c0, false, false);
    c1 = __builtin_amdgcn_wmma_f32_16x16x4_f32(false, a, false, b1, (short)0, c1, false, false);
    c2 = __builtin_amdgcn_wmma_f32_16x16x4_f32(false, a, false, b2, (short)0, c2, false, false);
    c3 = __builtin_amdgcn_wmma_f32_16x16x4_f32(false, a, false, b3, (short)0, c3, false, false);
  }
  // C/D layout: VGPR r holds row rbase+r, column n0 (+16/32/48 per tile).
  const int rbase = mtile * 16 + ((lane >> 4) << 3);
  float* __restrict__ yr = Y + (size_t)rbase * F + n0;
  if (rbase + 8 <= M) {  // interior tile: branch-free 32-store clause
#pragma unroll
    for (int r = 0; r < 8; ++r) {
      yr[r * F +  0] = c0[r];
      yr[r * F + 16] = c1[r];
      yr[r * F + 32] = c2[r];
      yr[r * F + 48] = c3[r];
    }
  } else {
#pragma unroll
    for (int r = 0; r < 8; ++r) {
      if (rbase + r < M) {
        yr[r * F +  0] = c0[r];
        yr[r * F + 16] = c1[r];
        yr[r * F + 32] = c2[r];
        yr[r * F + 48] = c3[r];
      }
    }
  }
}

static inline void launch_gemm(const float* X, const float* W, float* Y,
                               int M, int K, int F, hipStream_t stream) {
  dim3 grid((M + 63) / 64, F / 64);
  dim3 blk(128);
  if (K == 4 && F == 64) {
    gemm_wmma<4, 64><<<grid, blk, 0, stream>>>(X, W, Y, M);
  } else if (K == 64 && F == 128) {
    gemm_wmma<64, 128><<<grid, blk, 0, stream>>>(X, W, Y, M);
  } else if (K == 128 && F == 256) {
    gemm_wmma<128, 256><<<grid, blk, 0, stream>>>(X, W, Y, M);
  } else if (K == 768 && F == 256) {
    gemm_wmma<768, 256><<<grid, blk, 0, stream>>>(X, W, Y, M);
  }
}

// ---------------- per-column batch stats (biased), coalesced ----------------
// Block = (32,8): 32 consecutive columns, 8 row-threads; grid = (F/32, CHUNKS).
__global__ void col_partial(const float* __restrict__ Y, int M, int F,
                            float* __restrict__ sumb, float* __restrict__ sqb) {
  __shared__ float ss[8][32];
  __shared__ float sq[8][32];
  const int fx = blockIdx.x * 32 + threadIdx.x;
  const int nchunk = gridDim.y;
  const int rows_per = (M + nchunk - 1) / nchunk;
  const int r0 = blockIdx.y * rows_per;
  int r1 = r0 + rows_per;
  if (r1 > M) r1 = M;
  float s = 0.f, q = 0.f;
  for (int i = r0 + threadIdx.y; i < r1; i += 8) {
    float v = Y[(size_t)i * F + fx];
    s += v;
    q += v * v;
  }
  ss[threadIdx.y][threadIdx.x] = s;
  sq[threadIdx.y][threadIdx.x] = q;
  __syncthreads();
  if (threadIdx.y == 0) {
#pragma unroll
    for (int j = 1; j < 8; ++j) {
      s += ss[j][threadIdx.x];
      q += sq[j][threadIdx.x];
    }
    atomicAdd(&sumb[fx], s);
    atomicAdd(&sqb[fx], q);
  }
}

__global__ void finalize_stats(const float* __restrict__ sumb, const float* __restrict__ sqb,
                               float* __restrict__ mean, float* __restrict__ var, int M, int F) {
  int f = blockIdx.x * blockDim.x + threadIdx.x;
  if (f >= F) return;
  float im = 1.0f / (float)M;
  float m_ = sumb[f] * im;
  mean[f] = m_;
  var[f]  = sqb[f] * im - m_ * m_;
}

__global__ void bn_relu(const float* __restrict__ Y, const float* __restrict__ mean,
                        const float* __restrict__ var, const float* __restrict__ g,
                        const float* __restrict__ b, float* __restrict__ X,
                        size_t total, int F) {
  size_t i = (size_t)blockIdx.x * blockDim.x + threadIdx.x;
  if (i >= total) return;
  int f = (int)(i % (size_t)F);
  float v = (Y[i] - mean[f]) * rsqrtf(var[f] + 1e-5f) * g[f] + b[f];
  X[i] = v > 0.f ? v : 0.f;
}

// ---------------- alignment: gather scale-s features onto scale-1 voxels ----------------
__global__ void align_gather(const int4* __restrict__ uc1, const int* __restrict__ keys,
                             int Mk, const float* __restrict__ feat, float* __restrict__ cat,
                             int colOff, float vsx, float vsy, float vsz,
                             int shx, int shy, int shz) {
  const int row = blockIdx.x;
  __shared__ int s_idx;
  __shared__ int s_ok;
  if (threadIdx.x == 0) {
    int4 u = uc1[row];
    float rcx = ((float)u.y + 0.5f) * 0.2f;  // ref_center - pc_min
    float rcy = ((float)u.z + 0.5f) * 0.2f;
    float rcz = ((float)u.w + 0.5f) * 0.2f;
    int cx = (int)floorf(rcx / vsx);
    int cy = (int)floorf(rcy / vsy);
    int cz = (int)floorf(rcz / vsz);
    cx = cx < 0 ? 0 : (cx > shx - 1 ? shx - 1 : cx);
    cy = cy < 0 ? 0 : (cy > shy - 1 ? shy - 1 : cy);
    cz = cz < 0 ? 0 : (cz > shz - 1 ? shz - 1 : cz);
    long long k = (long long)u.x * 1000000000LL + (long long)cx * 1000000LL +
                  (long long)cy * 1000LL + (long long)cz;
    int lo = 0, hi = Mk;  // lower_bound == searchsorted(left)
    while (lo < hi) {
      int mid = (lo + hi) >> 1;
      if ((long long)keys[mid] < k) lo = mid + 1; else hi = mid;
    }
    int idx = lo;
    if (idx > Mk - 1) idx = Mk - 1;
    s_ok = ((long long)keys[idx] == k) ? 1 : 0;
    s_idx = idx;
  }
  __syncthreads();
  const int idx = s_idx;
  const int ok = s_ok;
  const int f = threadIdx.x;  // 256 features
  cat[(size_t)row * 768 + colOff + f] = ok ? feat[(size_t)idx * 256 + f] : 0.f;
}

__global__ void copy_center(const float* __restrict__ feat, float* __restrict__ cat, int M1) {
  size_t i = (size_t)blockIdx.x * blockDim.x + threadIdx.x;
  if (i >= (size_t)M1 * 256) return;
  size_t r = i >> 8;
  int f = (int)(i & 255);
  cat[r * 768 + 256 + f] = feat[i];
}

// ---------------- driver ----------------
static inline void do_stats(const float* Y, int M, int F, float* sumb, float* sqb,
                            float* meanb, float* varb, hipStream_t stream) {
  zero_f<<<2, 256, 0, stream>>>(sumb, (size_t)512);  // zero sumb+sqb (contiguous 512)
  col_partial<<<dim3(F / 32, 64), dim3(32, 8), 0, stream>>>(Y, M, F, sumb, sqb);
  finalize_stats<<<1, 256, 0, stream>>>(sumb, sqb, meanb, varb, M, F);
}

extern "C" void kernel_launch(void* const* d_in, const int* in_sizes, int n_in,
                              void* d_out, int out_size, void* d_ws, size_t ws_size,
                              hipStream_t stream) {
  const float* points = (const float*)d_in[0];
  const float* W1 = (const float*)d_in[1];
  const float* G1 = (const float*)d_in[2];
  const float* B1 = (const float*)d_in[3];
  const float* W2 = (const float*)d_in[4];
  const float* G2 = (const float*)d_in[5];
  const float* B2 = (const float*)d_in[6];
  const float* W3 = (const float*)d_in[7];
  const float* G3 = (const float*)d_in[8];
  const float* B3 = (const float*)d_in[9];
  const float* Wf = (const float*)d_in[10];
  const float* Gf = (const float*)d_in[11];
  const float* Bf = (const float*)d_in[12];
  const int* invp[3] = {(const int*)d_in[13], (const int*)d_in[14], (const int*)d_in[15]};
  const int* keys0 = (const int*)d_in[16];
  const int* keys2 = (const int*)d_in[17];
  const int* uc1   = (const int*)d_in[18];

  const int N  = in_sizes[13];
  const int M0 = in_sizes[16];
  const int M2 = in_sizes[17];
  const int M1 = in_sizes[18] / 4;
  const int Ms[3] = {M0, M1, M2};
  int Mmax = M0 > M1 ? M0 : M1;
  if (M2 > Mmax) Mmax = M2;

  // workspace carve (floats)
  float* tmpA  = (float*)d_ws;                // up to [Mmax, 256] (X / feat buffer)
  float* tmpB  = tmpA + (size_t)Mmax * 256;   // up to [Mmax, 256] (pre-BN GEMM out)
  float* cat   = tmpB + (size_t)Mmax * 256;   // [M1, 768]
  float* cnt   = cat + (size_t)M1 * 768;      // [Mmax]
  float* sumb  = cnt + (size_t)Mmax;          // [256]
  float* sqb   = sumb + 256;                  // [256]  (contiguous with sumb)
  float* meanb = sqb + 256;                   // [256]
  float* varb  = meanb + 256;                 // [256]

  for (int s = 0; s < 3; ++s) {
    const int M = Ms[s];
    const size_t z4 = (size_t)M * 4;
    zero_f<<<(unsigned)((z4 + 255) / 256), 256, 0, stream>>>(tmpB, z4);
    zero_f<<<(unsigned)((M + 255) / 256), 256, 0, stream>>>(cnt, (size_t)M);
    scatter_add<<<(N + 255) / 256, 256, 0, stream>>>(points, invp[s], tmpB, cnt, N);
    mean_div<<<(unsigned)((z4 + 255) / 256), 256, 0, stream>>>(tmpB, cnt, tmpA, M);

    // layer 1: 4 -> 64
    launch_gemm(tmpA, W1 + (size_t)s * 64 * 4, tmpB, M, 4, 64, stream);
    do_stats(tmpB, M, 64, sumb, sqb, meanb, varb, stream);
    bn_relu<<<(unsigned)(((size_t)M * 64 + 255) / 256), 256, 0, stream>>>(
        tmpB, meanb, varb, G1 + (size_t)s * 64, B1 + (size_t)s * 64, tmpA, (size_t)M * 64, 64);
    // layer 2: 64 -> 128
    launch_gemm(tmpA, W2 + (size_t)s * 128 * 64, tmpB, M, 64, 128, stream);
    do_stats(tmpB, M, 128, sumb, sqb, meanb, varb, stream);
    bn_relu<<<(unsigned)(((size_t)M * 128 + 255) / 256), 256, 0, stream>>>(
        tmpB, meanb, varb, G2 + (size_t)s * 128, B2 + (size_t)s * 128, tmpA, (size_t)M * 128, 128);
    // layer 3: 128 -> 256
    launch_gemm(tmpA, W3 + (size_t)s * 256 * 128, tmpB, M, 128, 256, stream);
    do_stats(tmpB, M, 256, sumb, sqb, meanb, varb, stream);
    bn_relu<<<(unsigned)(((size_t)M * 256 + 255) / 256), 256, 0, stream>>>(
        tmpB, meanb, varb, G3 + (size_t)s * 256, B3 + (size_t)s * 256, tmpA, (size_t)M * 256, 256);

    // route into concat buffer [M1, 768]
    if (s == 1) {
      copy_center<<<(unsigned)(((size_t)M1 * 256 + 255) / 256), 256, 0, stream>>>(tmpA, cat, M1);
    } else if (s == 0) {
      align_gather<<<M1, 256, 0, stream>>>((const int4*)uc1, keys0, M0, tmpA, cat,
                                           0, 0.1f, 0.1f, 0.1f, 1000, 1000, 60);
    } else {
      align_gather<<<M1, 256, 0, stream>>>((const int4*)uc1, keys2, M2, tmpA, cat,
                                           512, 0.4f, 0.4f, 0.4f, 250, 250, 15);
    }
  }

  // fused layer: 768 -> 256, then BN + ReLU into d_out
  launch_gemm(cat, Wf, tmpB, M1, 768, 256, stream);
  do_stats(tmpB, M1, 256, sumb, sqb, meanb, varb, stream);
  bn_relu<<<(unsigned)(((size_t)M1 * 256 + 255) / 256), 256, 0, stream>>>(
      tmpB, meanb, varb, Gf, Bf, (float*)d_out, (size_t)M1 * 256, 256);
}